// ScaledDotProduct_49314814493101
// MI455X (gfx1250) — compile-verified
//
#include <hip/hip_runtime.h>
#include <hip/hip_bf16.h>
#include <math.h>

// ---- problem constants (from reference) ----
#define BB       4
#define SS       8
#define CC       1024
#define DKK      1024
#define N3       (3 * DKK)
#define MAXLEN   8192
#define TBLK     128          // rows per prefix block
#define NTC      64           // channels per k_values workgroup (4 waves x 16)
#define SLAB     (TBLK * 16 + 16)   // padded per-wave LDS slab (floats)

typedef __attribute__((ext_vector_type(2))) float v2f;
typedef __attribute__((ext_vector_type(8))) float v8f;

#define GLOBAL_AS __attribute__((address_space(1)))
#define LDS_AS    __attribute__((address_space(3)))

#if defined(__has_builtin)
#if __has_builtin(__builtin_amdgcn_global_load_async_to_lds_b32)
#define ASYNC_STAGE 1
#endif
#endif

static __device__ __forceinline__ void wait_async0() {
#if defined(__has_builtin) && __has_builtin(__builtin_amdgcn_s_wait_asynccnt)
    __builtin_amdgcn_s_wait_asynccnt(0);
#else
    asm volatile("s_wait_asynccnt 0x0" ::: "memory");
#endif
}

// ------------------------------------------------------------------
// K1: qkv = x_last @ W + b   (A padded 16 x 1024, B = W column tile)
// one wave per 16-column tile of the 3072 outputs
// ------------------------------------------------------------------
__global__ void k_qkv(const float* __restrict__ x, const float* __restrict__ W,
                      const float* __restrict__ bias, float* __restrict__ qkv)
{
    const int n0    = blockIdx.x * 16;
    const int lane  = threadIdx.x;        // 0..31
    const int rowA  = lane & 15;          // M index for this lane's A values
    const int kHalf = (lane >> 4) * 2;    // lanes 0-15: K={0,1}; lanes 16-31: K={2,3}
    const int col   = lane & 15;          // N index for this lane's B values

    // A row = x[b, S-1, :] for b<4, else zero (padded). Always load valid memory.
    const int   rclamp = rowA < (BB - 1) ? rowA : (BB - 1);
    const float mA     = (rowA < BB) ? 1.0f : 0.0f;
    const float* xrow  = x + ((size_t)rclamp * SS + (SS - 1)) * CC;

    v8f acc = {};
    for (int k0 = 0; k0 < CC; k0 += 4) {
        v2f a, bf;
        a.x  = xrow[k0 + kHalf + 0] * mA;
        a.y  = xrow[k0 + kHalf + 1] * mA;
        bf.x = W[(size_t)(k0 + kHalf + 0) * N3 + n0 + col];
        bf.y = W[(size_t)(k0 + kHalf + 1) * N3 + n0 + col];
        acc = __builtin_amdgcn_wmma_f32_16x16x4_f32(false, a, false, bf,
                                                    (short)0, acc, false, false);
    }
    // D layout: VGPR r -> row r (lanes 0-15). Only rows 0..3 are real batches.
    if (lane < 16) {
        #pragma unroll
        for (int r = 0; r < BB; ++r)
            qkv[(size_t)r * N3 + n0 + lane] = acc[r] + bias[n0 + lane];
    }
}

// ------------------------------------------------------------------
// K2: w[b][T] = dot(q[b], key_row(T)) * scale ; row size-1 comes from k_new
// one wave per (b,T)
// ------------------------------------------------------------------
__global__ void k_scores(const float* __restrict__ qkv,
                         const float* __restrict__ key_cache,
                         float* __restrict__ w, int size, float scale)
{
    const int wid  = blockIdx.x * (blockDim.x >> 5) + (threadIdx.x >> 5);
    const int lane = threadIdx.x & 31;
    const int b    = wid / size;
    const int T    = wid - b * size;
    if (b >= BB) return;

    const float* q = qkv + (size_t)b * N3;                       // q = cols [0,DK)
    const float* krow = (T == size - 1)
                      ? (qkv + (size_t)b * N3 + DKK)             // k_new
                      : (key_cache + ((size_t)b * MAXLEN + T) * DKK);

    float s = 0.0f;
    for (int c = lane; c < DKK; c += 32) s += q[c] * krow[c];
    #pragma unroll
    for (int off = 16; off >= 1; off >>= 1) s += __shfl_xor(s, off, 32);
    if (lane == 0) w[(size_t)b * size + T] = s * scale;
}

// ------------------------------------------------------------------
// K3: per batch: m = max(w), e = exp(w-m), Z = inclusive prefix sum of e
// ------------------------------------------------------------------
__global__ void k_scan(const float* __restrict__ w, float* __restrict__ e,
                       float* __restrict__ Z, int size)
{
    __shared__ float red[1024];
    const int b    = blockIdx.x;
    const int tid  = threadIdx.x;
    const int nthr = blockDim.x;
    const int nPer = size / nthr;                 // 4 for size=4096
    const float* wb = w + (size_t)b * size;
    float*       eb = e + (size_t)b * size;
    float*       Zb = Z + (size_t)b * size;

    float lm = -1e30f;
    for (int i = tid; i < size; i += nthr) lm = fmaxf(lm, wb[i]);
    red[tid] = lm; __syncthreads();
    for (int s = nthr >> 1; s >= 1; s >>= 1) {
        if (tid < s) red[tid] = fmaxf(red[tid], red[tid + s]);
        __syncthreads();
    }
    const float m = red[0];
    __syncthreads();

    float run = 0.0f;
    for (int i = 0; i < nPer; ++i) {
        float ex = __expf(wb[tid * nPer + i] - m);
        eb[tid * nPer + i] = ex;
        run += ex;
    }
    red[tid] = run; __syncthreads();

    for (int s = 1; s < nthr; s <<= 1) {
        float t = (tid >= s) ? red[tid - s] : 0.0f;
        __syncthreads();
        red[tid] += t;
        __syncthreads();
    }
    float acc = red[tid] - run;                   // exclusive prefix for this chunk
    for (int i = 0; i < nPer; ++i) {
        acc += eb[tid * nPer + i];
        Zb[tid * nPer + i] = acc;
    }
}

// ------------------------------------------------------------------
// K4: block partials  P[b][j][c] = sum_{T in block j} e[T] * v_row(T)[c]
// ------------------------------------------------------------------
__global__ void k_blocksum(const float* __restrict__ e,
                           const float* __restrict__ value_cache,
                           const float* __restrict__ qkv,
                           float* __restrict__ P, int size, int nblocks)
{
    const int b  = blockIdx.x / nblocks;
    const int j  = blockIdx.x - b * nblocks;
    const int c  = threadIdx.x;
    const int t0 = j * TBLK;
    const float* eb = e + (size_t)b * size;
    const float* vnew = qkv + (size_t)b * N3 + 2 * DKK;

    float s = 0.0f;
    for (int r = 0; r < TBLK; ++r) {
        const int T = t0 + r;
        const float* vrow = (T == size - 1)
                          ? vnew
                          : (value_cache + ((size_t)b * MAXLEN + T) * DKK);
        __builtin_prefetch(vrow + DKK, 0, 1);     // global_prefetch_b8 (gfx1250)
        s += eb[T] * vrow[c];
    }
    P[((size_t)b * nblocks + j) * DKK + c] = s;
}

// ------------------------------------------------------------------
// K5: in-place exclusive prefix of P over the block axis (per b,c)
// ------------------------------------------------------------------
__global__ void k_blockprefix(float* __restrict__ P, int nblocks)
{
    const int idx = blockIdx.x * blockDim.x + threadIdx.x;   // over B*DK columns
    const int b   = idx / DKK;
    const int c   = idx - b * DKK;
    if (b >= BB) return;
    float run = 0.0f;
    for (int j = 0; j < nblocks; ++j) {
        const size_t o = ((size_t)b * nblocks + j) * DKK + c;
        const float p = P[o];
        P[o] = run;                                          // exclusive
        run += p;
    }
}

// ------------------------------------------------------------------
// K6: out[b][t][c] = (base + within-block prefix of e*v) / Z[t]
// 4 waves / 64 channels per workgroup. V block staged to LDS with
// async global->LDS (ASYNCcnt) when available; each wave owns a
// padded 16-column slab and runs triangular v_wmma_f32_16x16x4_f32.
// ------------------------------------------------------------------
__global__ void k_values(const float* __restrict__ e, const float* __restrict__ Z,
                         const float* __restrict__ base,
                         const float* __restrict__ value_cache,
                         const float* __restrict__ qkv,
                         float* __restrict__ out, int size, int nblocks)
{
    __shared__ float EV[4 * SLAB];                // ~33 KB
    const int ctiles = DKK / NTC;                 // 16
    const int wg   = blockIdx.x;
    const int n0   = (wg % ctiles) * NTC;         // 64-channel tile base
    const int j    = (wg / ctiles) % nblocks;
    const int b    = wg / (ctiles * nblocks);
    const int tid  = threadIdx.x;                 // 0..127
    const int lane = tid & 31;
    const int wv   = tid >> 5;                    // wave id 0..3
    const int t0   = j * TBLK;

    const float* eb   = e + (size_t)b * size;
    const float* Zb   = Z + (size_t)b * size;
    const float* vnew = qkv + (size_t)b * N3 + 2 * DKK;

    // ---- stage raw V block (128 rows x 64 chans) into padded LDS slabs ----
    for (int idx = tid; idx < TBLK * NTC; idx += blockDim.x) {
        const int r  = idx >> 6;                  // row 0..127
        const int c  = idx & 63;                  // channel 0..63
        const int T  = t0 + r;
        const float* vrow = (T == size - 1)
                          ? vnew
                          : (value_cache + ((size_t)b * MAXLEN + T) * DKK);
        const int daddr = (c >> 4) * SLAB + r * 16 + (c & 15);
#if defined(ASYNC_STAGE)
        __builtin_amdgcn_global_load_async_to_lds_b32(
            (GLOBAL_AS int*)(vrow + n0 + c), (LDS_AS int*)&EV[daddr], 0, 0);
#else
        EV[daddr] = vrow[n0 + c];
#endif
    }
#if defined(ASYNC_STAGE)
    wait_async0();
#endif
    __syncthreads();

    // ---- scale by e in LDS ----
    for (int idx = tid; idx < TBLK * NTC; idx += blockDim.x) {
        const int r = idx >> 6;
        const int c = idx & 63;
        const int daddr = (c >> 4) * SLAB + r * 16 + (c & 15);
        EV[daddr] *= eb[t0 + r];
    }
    __syncthreads();

    // ---- per-wave triangular prefix on its 16-column slab ----
    float* evb = EV + wv * SLAB;
    const int col   = lane & 15;
    const int rowA  = lane & 15;
    const int kHalf = (lane >> 4) * 2;
    const int rbase = (lane >> 4) * 8;            // D rows 0..7 / 8..15 by lane half
    const int cgl   = n0 + wv * 16 + col;         // this lane's output channel

    float runbase = base[((size_t)b * nblocks + j) * DKK + cgl];

    for (int i = 0; i < TBLK / 16; ++i) {
        v8f acc = {};
        #pragma unroll
        for (int k0 = 0; k0 < 16; k0 += 4) {
            v2f a, bf;
            // A = 16x16 lower-triangular ones (inclusive diagonal)
            a.x = (rowA >= k0 + kHalf + 0) ? 1.0f : 0.0f;
            a.y = (rowA >= k0 + kHalf + 1) ? 1.0f : 0.0f;
            // B = EV rows (i*16 + k), this lane's column
            bf.x = evb[(i * 16 + k0 + kHalf + 0) * 16 + col];
            bf.y = evb[(i * 16 + k0 + kHalf + 1) * 16 + col];
            acc = __builtin_amdgcn_wmma_f32_16x16x4_f32(false, a, false, bf,
                                                        (short)0, acc, false, false);
        }
        #pragma unroll
        for (int r = 0; r < 8; ++r) {
            const int t = t0 + i * 16 + rbase + r;
            const float numer = runbase + acc[r];
            out[((size_t)b * size + t) * DKK + cgl] = numer / Zb[t];
        }
        float cs = 0.0f;
        #pragma unroll
        for (int r = 0; r < 16; ++r) cs += evb[(i * 16 + r) * 16 + col];
        runbase += cs;
    }
}

// ------------------------------------------------------------------
extern "C" void kernel_launch(void* const* d_in, const int* in_sizes, int n_in,
                              void* d_out, int out_size, void* d_ws, size_t ws_size,
                              hipStream_t stream)
{
    const float* x           = (const float*)d_in[0];
    const float* W           = (const float*)d_in[1];
    const float* bias        = (const float*)d_in[2];
    const float* key_cache   = (const float*)d_in[3];
    const float* value_cache = (const float*)d_in[4];
    (void)in_sizes; (void)n_in; (void)ws_size;

    const int size = out_size / (BB * DKK);       // = token_ind + 1 = 4096
    if (size <= 0) return;
    const int nblocks = size / TBLK;              // 32

    float* ws  = (float*)d_ws;
    float* qkv = ws;                              // B * 3DK
    float* w   = qkv + (size_t)BB * N3;           // B * size
    float* e   = w   + (size_t)BB * size;         // B * size
    float* Z   = e   + (size_t)BB * size;         // B * size
    float* P   = Z   + (size_t)BB * size;         // B * nblocks * DK

    const float scale = 1.0f / sqrtf((float)DKK);

    k_qkv        <<<N3 / 16,            32,   0, stream>>>(x, W, bias, qkv);
    k_scores     <<<(BB * size) / 8,    256,  0, stream>>>(qkv, key_cache, w, size, scale);
    k_scan       <<<BB,                 1024, 0, stream>>>(w, e, Z, size);
    k_blocksum   <<<BB * nblocks,       1024, 0, stream>>>(e, value_cache, qkv, P, size, nblocks);
    k_blockprefix<<<(BB * DKK) / 256,   256,  0, stream>>>(P, nblocks);
    k_values     <<<BB * nblocks * (DKK / NTC), 128, 0, stream>>>(e, Z, P, value_cache, qkv,
                                                                  (float*)d_out, size, nblocks);
}